// FusionGNN_16484084483814
// MI455X (gfx1250) — compile-verified
//
#include <hip/hip_runtime.h>
#include <hip/hip_bf16.h>

// ---------------------------------------------------------------------------
// FusionGNN forward for MI455X (gfx1250), fp32 compute with
// V_WMMA_F32_16X16X4_F32 tensor ops (reference is fp32; BN stats / softmax
// are precision sensitive, so no bf16 downcast).
// All weights are pre-padded into workspace so GEMM inner loops have ZERO
// conditional loads; fragment loads are double-buffered against the WMMAs.
// ---------------------------------------------------------------------------

typedef float v2f __attribute__((ext_vector_type(2)));
typedef float v8f __attribute__((ext_vector_type(8)));

#define NN 131072
#define EE 262144
#define BG 4096
#define DD 300
#define DP 304      // padded leading dim for D=300 (19 tiles of 16)
#define DW 320      // padded GEMM width for 300 (5 groups of 64)
#define D2 600
#define D2P 608     // padded hidden width (38 tiles of 16)
#define FDIM 512
#define CDIM 908
#define ZDIM 812
#define ZS 306      // LDS stride for z tile   (bank-conflict free, 8B aligned)
#define HS 612      // LDS stride for hidden   (bank-conflict free, 8B aligned)

#define FL_RELU 1
#define FL_BIAS 2
#define FL_GATH 4

// D = A(16x4) * B(4x16) + C ; fp32 WMMA, wave32
__device__ __forceinline__ v8f wmma4(v2f a, v2f b, v8f c) {
  return __builtin_amdgcn_wmma_f32_16x16x4_f32(false, a, false, b, (short)0, c,
                                               false, false);
}

// ---------------------------------------------------------------------------
// Generic fp32 WMMA GEMM:  O[M,n] = f(A[M,K] @ W[K,np] (+bias) (+gsrc[gidx[r]]))
// W must be padded: ldw == np == gridDim.x*64 (zero-filled cols >= n).
// block = 128 threads (4 waves); wave w -> rows by*64+w*16; 4 col tiles/wave.
// M multiple of 64; K multiple of 4. Loads are unconditional; only the store
// epilogue guards col < n. Fragment loads double-buffered.
// ---------------------------------------------------------------------------
__global__ void k_gemm(const float* __restrict__ A, int lda,
                       const float* __restrict__ W, int ldw,
                       const float* __restrict__ bias,
                       float* __restrict__ O, int ldo,
                       int K, int n, int flags,
                       const int* __restrict__ gidx,
                       const float* __restrict__ gsrc, int ldg) {
  const int lane = threadIdx.x & 31;
  const int wv = threadIdx.x >> 5;
  const int m0 = blockIdx.y * 64 + wv * 16;
  const int cb = blockIdx.x * 64;
  const int row = m0 + (lane & 15);
  const int koff = (lane >= 16) ? 2 : 0;
  const int colL = lane & 15;
  const int rsel = (lane >= 16) ? 8 : 0;

  const float* Ap = A + (long)row * lda + koff;
  const float* Wp = W + (long)koff * ldw + cb + colL;

  v8f acc0 = (v8f){0.f,0.f,0.f,0.f,0.f,0.f,0.f,0.f};
  v8f acc1 = acc0, acc2 = acc0, acc3 = acc0;

  // prologue fragments (k = 0)
  v2f a_c;  a_c.x = Ap[0]; a_c.y = Ap[1];
  v2f b_c0, b_c1, b_c2, b_c3;
  b_c0.x = Wp[0];  b_c0.y = Wp[ldw + 0];
  b_c1.x = Wp[16]; b_c1.y = Wp[ldw + 16];
  b_c2.x = Wp[32]; b_c2.y = Wp[ldw + 32];
  b_c3.x = Wp[48]; b_c3.y = Wp[ldw + 48];

  for (int k = 4; k < K; k += 4) {
    // issue next-step loads first, then do this step's WMMAs
    const float* Wk = Wp + (long)k * ldw;
    v2f a_n;  a_n.x = Ap[k]; a_n.y = Ap[k + 1];
    v2f b_n0, b_n1, b_n2, b_n3;
    b_n0.x = Wk[0];  b_n0.y = Wk[ldw + 0];
    b_n1.x = Wk[16]; b_n1.y = Wk[ldw + 16];
    b_n2.x = Wk[32]; b_n2.y = Wk[ldw + 32];
    b_n3.x = Wk[48]; b_n3.y = Wk[ldw + 48];
    acc0 = wmma4(a_c, b_c0, acc0);
    acc1 = wmma4(a_c, b_c1, acc1);
    acc2 = wmma4(a_c, b_c2, acc2);
    acc3 = wmma4(a_c, b_c3, acc3);
    a_c = a_n;
    b_c0 = b_n0; b_c1 = b_n1; b_c2 = b_n2; b_c3 = b_n3;
  }
  acc0 = wmma4(a_c, b_c0, acc0);
  acc1 = wmma4(a_c, b_c1, acc1);
  acc2 = wmma4(a_c, b_c2, acc2);
  acc3 = wmma4(a_c, b_c3, acc3);

  v8f accs[4] = {acc0, acc1, acc2, acc3};
#pragma unroll
  for (int t = 0; t < 4; ++t) {
    int col = cb + t * 16 + colL;
    if (col >= n) continue;
    float bv = (flags & FL_BIAS) ? bias[col] : 0.f;
#pragma unroll
    for (int v = 0; v < 8; ++v) {
      int r = m0 + rsel + v;
      float x = accs[t][v] + bv;
      if (flags & FL_GATH) x += gsrc[(long)gidx[r] * ldg + col];
      if (flags & FL_RELU) x = fmaxf(x, 0.f);
      O[(long)r * ldo + col] = x;
    }
  }
}

// ---------------------------------------------------------------------------
// Fused GINE MLP: out[16 rows] = relu((h+agg) @ W1p + b1p) @ W2p + b2p
// W1p: 300 x 608 (zero-padded), W2p: 600 x 304 (zero-padded); b1p/b2p padded.
// One WG (256 thr / 8 waves) per 16-row strip; hidden lives in LDS.
// Safe with out == h (each WG reads only its own rows before writing them).
// ---------------------------------------------------------------------------
__global__ void k_gine(const float* h, const float* agg,
                       const float* __restrict__ W1, const float* __restrict__ b1,
                       const float* __restrict__ W2, const float* __restrict__ b2,
                       float* out) {
  __shared__ float sh_z[16 * ZS];
  __shared__ float sh_hid[16 * HS];
  const int tid = threadIdx.x;
  const int lane = tid & 31;
  const int wv = tid >> 5;
  const long m0 = (long)blockIdx.x * 16;
  const int koff = (lane >= 16) ? 2 : 0;
  const int colL = lane & 15;
  const int rsel = (lane >= 16) ? 8 : 0;
  const int rA = lane & 15;

  // stage 0: z = h + agg into LDS
  for (int i = tid; i < 16 * DP; i += 256) {
    int r = i / DP;
    int c = i - r * DP;
    float v = 0.f;
    if (c < DD) v = h[(m0 + r) * DP + c] + agg[(m0 + r) * DP + c];
    sh_z[r * ZS + c] = v;
  }
  __syncthreads();

  // stage 1: hidden = relu(z @ W1 + b1), 38 col tiles of 16 (608 padded cols)
  for (int t = wv; t < 38; t += 8) {
    v8f acc = (v8f){0.f,0.f,0.f,0.f,0.f,0.f,0.f,0.f};
    const int col = t * 16 + colL;
    const float* Wp = W1 + (long)koff * D2P + col;
    v2f a_c, b_c;
    a_c.x = sh_z[rA * ZS + koff];
    a_c.y = sh_z[rA * ZS + koff + 1];
    b_c.x = Wp[0];
    b_c.y = Wp[D2P];
    for (int k = 4; k < DD; k += 4) {
      v2f a_n, b_n;
      a_n.x = sh_z[rA * ZS + k + koff];
      a_n.y = sh_z[rA * ZS + k + koff + 1];
      b_n.x = Wp[(long)k * D2P];
      b_n.y = Wp[(long)(k + 1) * D2P];
      acc = wmma4(a_c, b_c, acc);
      a_c = a_n; b_c = b_n;
    }
    acc = wmma4(a_c, b_c, acc);
    float bv = b1[col];
#pragma unroll
    for (int v = 0; v < 8; ++v)
      sh_hid[(v + rsel) * HS + col] = fmaxf(acc[v] + bv, 0.f);
  }
  __syncthreads();

  // stage 2: out = hidden @ W2 + b2, 19 col tiles (304 padded cols)
  for (int t = wv; t < 19; t += 8) {
    v8f acc = (v8f){0.f,0.f,0.f,0.f,0.f,0.f,0.f,0.f};
    const int col = t * 16 + colL;
    const float* Wp = W2 + (long)koff * DP + col;
    v2f a_c, b_c;
    a_c.x = sh_hid[rA * HS + koff];
    a_c.y = sh_hid[rA * HS + koff + 1];
    b_c.x = Wp[0];
    b_c.y = Wp[DP];
    for (int k = 4; k < D2; k += 4) {
      v2f a_n, b_n;
      a_n.x = sh_hid[rA * HS + k + koff];
      a_n.y = sh_hid[rA * HS + k + koff + 1];
      b_n.x = Wp[(long)k * DP];
      b_n.y = Wp[(long)(k + 1) * DP];
      acc = wmma4(a_c, b_c, acc);
      a_c = a_n; b_c = b_n;
    }
    acc = wmma4(a_c, b_c, acc);
    float bv = b2[col];
#pragma unroll
    for (int v = 0; v < 8; ++v)
      out[(m0 + v + rsel) * DP + col] = acc[v] + bv;  // pad cols get 0
  }
}

// ---------------------------------------------------------------------------
// Elementwise / scatter / reduction helpers
// ---------------------------------------------------------------------------
__global__ void k_zero(float* p, long n) {
  long i = (long)blockIdx.x * blockDim.x + threadIdx.x;
  if (i < n) p[i] = 0.f;
}

// pad src[rows, cols] row-major into dst[rows, ldd], zero-filling cols >= cols
__global__ void k_padw(const float* __restrict__ src, int rows, int cols,
                       float* __restrict__ dst, int ldd) {
  long i = (long)blockIdx.x * blockDim.x + threadIdx.x;
  if (i >= (long)rows * ldd) return;
  int r = (int)(i / ldd);
  int c = (int)(i - (long)r * ldd);
  dst[i] = (c < cols) ? src[(long)r * cols + c] : 0.f;
}

__global__ void k_embed(const int* __restrict__ x, const float* __restrict__ emb,
                        float* __restrict__ h, int rows, int xs) {
  long i = (long)blockIdx.x * blockDim.x + threadIdx.x;
  if (i >= (long)rows * DD) return;
  int r = (int)(i / DD);
  int c = (int)(i - (long)r * DD);
  h[(long)r * DP + c] = emb[(long)x[(long)r * xs] * DD + c];
}

// agg[dst] += relu(h[src] + edge_emb[ea0]); 4 cols per thread (75 chunks/edge)
__global__ void k_scatter(const int* __restrict__ ei, const int* __restrict__ ea,
                          const float* __restrict__ h, const float* __restrict__ emb,
                          float* agg) {
  int idx = blockIdx.x * blockDim.x + threadIdx.x;
  if (idx >= EE * 75) return;
  int e = idx / 75;
  int c0 = (idx - e * 75) * 4;
  int src = ei[e];
  int dst = ei[EE + e];
  int a = ea[2 * e];
  const float4 hv = *(const float4*)(h + (long)src * DP + c0);
  const float4 ev = *(const float4*)(emb + (long)a * DD + c0);
  float m0 = fmaxf(hv.x + ev.x, 0.f), m1 = fmaxf(hv.y + ev.y, 0.f);
  float m2 = fmaxf(hv.z + ev.z, 0.f), m3 = fmaxf(hv.w + ev.w, 0.f);
  float* base = agg + (long)dst * DP + c0;
  atomicAdd(base + 0, m0);
  atomicAdd(base + 1, m1);
  atomicAdd(base + 2, m2);
  atomicAdd(base + 3, m3);
}

__global__ void k_bnstats(const float* __restrict__ x, float* sum, float* sumsq,
                          int rowsPerBlk) {
  int c1 = threadIdx.x;          // 0..255 (<300)
  int c2 = c1 + 256;
  bool has2 = c2 < DD;
  long r0 = (long)blockIdx.x * rowsPerBlk;
  float s1 = 0.f, q1 = 0.f, s2 = 0.f, q2 = 0.f;
  for (int r = 0; r < rowsPerBlk; ++r) {
    const float* rowp = x + (r0 + r) * DP;
    float v = rowp[c1];
    s1 += v; q1 += v * v;
    if (has2) { float w = rowp[c2]; s2 += w; q2 += w * w; }
  }
  atomicAdd(&sum[c1], s1);
  atomicAdd(&sumsq[c1], q1);
  if (has2) { atomicAdd(&sum[c2], s2); atomicAdd(&sumsq[c2], q2); }
}

__global__ void k_bnapply(const float* __restrict__ x, const float* __restrict__ sum,
                          const float* __restrict__ sumsq, const float* __restrict__ g,
                          const float* __restrict__ b, float* o, int relu) {
  long i = (long)blockIdx.x * blockDim.x + threadIdx.x;
  if (i >= (long)NN * DD) return;
  long r = i / DD;
  int c = (int)(i - r * DD);
  const float inv_n = 1.f / (float)NN;
  float mean = sum[c] * inv_n;
  float var = sumsq[c] * inv_n - mean * mean;
  float inv = rsqrtf(var + 1e-5f);
  float v = (x[r * DP + c] - mean) * inv * g[c] + b[c];
  if (relu) v = fmaxf(v, 0.f);
  o[r * DP + c] = v;
}

// wvec[k] = dot(Wdst[k,:], adst)   (constant across layers)
__global__ void k_wvec(const float* __restrict__ Wdst, const float* __restrict__ adst,
                       float* w) {
  int k = blockIdx.x * blockDim.x + threadIdx.x;
  if (k >= DD) return;
  float s = 0.f;
  for (int j = 0; j < DD; ++j) s += Wdst[(long)k * DD + j] * adst[j];
  w[k] = s;
}

// t[b] = dot(ctx[b,:300], wvec)
__global__ void k_tctx(const float* __restrict__ ctx, const float* __restrict__ w,
                       float* t) {
  int b = blockIdx.x * blockDim.x + threadIdx.x;
  if (b >= BG) return;
  float s = 0.f;
  for (int k = 0; k < DD; ++k) s += ctx[(long)b * DP + k] * w[k];
  t[b] = s;
}

// per node: ex[i] = exp(leaky(hs[i].asrc + t[batch[i]])); den[b] += ex
__global__ void k_logits(const float* __restrict__ hs, const float* __restrict__ asrc,
                         const float* __restrict__ t, const int* __restrict__ batch,
                         float* ex, float* den) {
  int lane = threadIdx.x & 31;
  int wv = threadIdx.x >> 5;
  long i = (long)blockIdx.x * 8 + wv;
  float p = 0.f;
  for (int k = lane; k < DD; k += 32) p += hs[i * DP + k] * asrc[k];
#pragma unroll
  for (int off = 16; off; off >>= 1) p += __shfl_xor(p, off, 32);
  if (lane == 0) {
    int b = batch[i];
    float l = p + t[b];
    l = (l > 0.f) ? l : 0.2f * l;
    float e = expf(l);
    ex[i] = e;
    atomicAdd(&den[b], e);
  }
}

__global__ void k_alpha(float* ex, const float* __restrict__ den,
                        const int* __restrict__ batch) {
  long i = (long)blockIdx.x * blockDim.x + threadIdx.x;
  if (i >= NN) return;
  ex[i] = ex[i] / den[batch[i]];
}

__global__ void k_gatacc(const float* __restrict__ hs, const float* __restrict__ alpha,
                         const int* __restrict__ batch, float* acc) {
  long i = (long)blockIdx.x * blockDim.x + threadIdx.x;
  if (i >= (long)NN * DD) return;
  long r = i / DD;
  int c = (int)(i - r * DD);
  atomicAdd(&acc[(long)batch[r] * DP + c], alpha[r] * hs[r * DP + c]);
}

__global__ void k_ctxcomb(const float* __restrict__ aA, const float* __restrict__ aB,
                          const float* __restrict__ bias, float* ctx) {
  long i = (long)blockIdx.x * blockDim.x + threadIdx.x;
  if (i >= (long)BG * DD) return;
  long b = i / DD;
  int c = (int)(i - b * DD);
  ctx[b * DP + c] = 0.5f * (aA[b * DP + c] + aB[b * DP + c]) + bias[c];
}

__global__ void k_poolacc(const float* __restrict__ h, const int* __restrict__ batch,
                          float* pool) {
  long i = (long)blockIdx.x * blockDim.x + threadIdx.x;
  if (i >= (long)NN * DD) return;
  long r = i / DD;
  int c = (int)(i - r * DD);
  atomicAdd(&pool[(long)batch[r] * DP + c], h[r * DP + c]);
}

__global__ void k_count(const int* __restrict__ batch, float* cnt) {
  long i = (long)blockIdx.x * blockDim.x + threadIdx.x;
  if (i >= NN) return;
  atomicAdd(&cnt[batch[i]], 1.f);
}

__global__ void k_pooldiv(float* pool, const float* __restrict__ cnt) {
  long i = (long)blockIdx.x * blockDim.x + threadIdx.x;
  if (i >= (long)BG * DD) return;
  long b = i / DD;
  int c = (int)(i - b * DD);
  pool[b * DP + c] /= fmaxf(cnt[b], 1.f);
}

__global__ void k_copy2d(const float* __restrict__ in, int ldi, float* out, int ldo,
                         int off, int ncols) {
  long i = (long)blockIdx.x * blockDim.x + threadIdx.x;
  if (i >= (long)BG * ncols) return;
  long b = i / ncols;
  int c = (int)(i - b * ncols);
  out[b * ldo + off + c] = in[b * ldi + c];
}

__global__ void k_final(const float* __restrict__ m2, const float* __restrict__ W3,
                        const float* __restrict__ b3, float* out) {
  int b = blockIdx.x * blockDim.x + threadIdx.x;
  if (b >= BG) return;
  float s = b3[0];
  for (int j = 0; j < 64; ++j) s += m2[(long)b * 64 + j] * W3[j];
  out[b] = s;
}

// ---------------------------------------------------------------------------
extern "C" void kernel_launch(void* const* d_in, const int* in_sizes, int n_in,
                              void* d_out, int out_size, void* d_ws, size_t ws_size,
                              hipStream_t stream) {
  (void)in_sizes; (void)n_in; (void)out_size; (void)ws_size;

  const int* xA  = (const int*)d_in[0];
  const int* eiA = (const int*)d_in[1];
  const int* eaA = (const int*)d_in[2];
  const int* bA  = (const int*)d_in[3];
  const int* xB  = (const int*)d_in[4];
  const int* eiB = (const int*)d_in[5];
  const int* eaB = (const int*)d_in[6];
  const int* bB  = (const int*)d_in[7];
  const float* context = (const float*)d_in[8];
  // params: dict insertion order
  const float* x_emb1   = (const float*)d_in[9];
  const float* edge_emb1= (const float*)d_in[10];
  const float* gine_W1  = (const float*)d_in[11];
  const float* gine_b1  = (const float*)d_in[12];
  const float* gine_W2  = (const float*)d_in[13];
  const float* gine_b2  = (const float*)d_in[14];
  const float* bn_g     = (const float*)d_in[15];
  const float* bn_b     = (const float*)d_in[16];
  const float* sage_Wl  = (const float*)d_in[17];
  const float* sage_bl  = (const float*)d_in[18];
  const float* sage_Wr  = (const float*)d_in[19];
  const float* gat_Wsrc = (const float*)d_in[20];
  const float* gat_Wdst = (const float*)d_in[21];
  const float* gat_asrc = (const float*)d_in[22];
  const float* gat_adst = (const float*)d_in[23];
  const float* gat_bias = (const float*)d_in[24];
  const float* ce_W1    = (const float*)d_in[25];
  const float* ce_b1    = (const float*)d_in[26];
  const float* ce_W2    = (const float*)d_in[27];
  const float* ce_b2    = (const float*)d_in[28];
  const float* ol_W1    = (const float*)d_in[29];
  const float* ol_b1    = (const float*)d_in[30];
  const float* ol_W2    = (const float*)d_in[31];
  const float* ol_b2    = (const float*)d_in[32];
  const float* mlp_W1   = (const float*)d_in[33];
  const float* mlp_b1   = (const float*)d_in[34];
  const float* mlp_W2   = (const float*)d_in[35];
  const float* mlp_b2   = (const float*)d_in[36];
  const float* mlp_W3   = (const float*)d_in[37];
  const float* mlp_b3   = (const float*)d_in[38];

  // ---- workspace layout (floats) ----
  float* w = (float*)d_ws;
  const size_t NP = (size_t)NN * DP;
  const size_t BP = (size_t)BG * DP;
  auto take = [&](size_t n) { float* p = w; w += n; return p; };
  float* hA    = take(NP);
  float* hB    = take(NP);
  float* tmp1  = take(NP);   // inject output / GAT hs
  float* tmp2  = take(NP);   // agg / GAT alpha
  float* ctx0  = take(BP);
  float* ctx1  = take(BP);
  float* sageL = take(BP);
  float* accA  = take(BP);
  float* accB  = take(BP);
  float* poolA = take(BP);
  float* poolB = take(BP);
  float* ceH   = take((size_t)BG * FDIM);  // also reused as olH
  float* gAo   = take((size_t)BG * 256);
  float* gBo   = take((size_t)BG * 256);
  float* zcat  = take((size_t)BG * ZDIM);
  float* m1    = take((size_t)BG * 256);
  float* m2    = take((size_t)BG * 64);
  float* sumb  = take(DP);
  float* sumsq = take(DP);
  float* wvec  = take(DP);
  float* tvec  = take(BG);
  float* cntA  = take(BG);
  float* cntB  = take(BG);
  float* den   = take(BG);
  // padded weights (zero-filled columns -> guard-free GEMM inner loops)
  float* gW1p  = take((size_t)5 * DD * D2P);   // (5*300) x 608
  float* gW2p  = take((size_t)5 * D2 * DP);    // (5*600) x 304
  float* gb1p  = take((size_t)5 * D2P);
  float* gb2p  = take((size_t)5 * DP);
  float* ceW2p = take((size_t)FDIM * DW);      // 512 x 320
  float* sWlp  = take((size_t)DD * DW);        // 300 x 320
  float* sWrp  = take((size_t)DD * DW);
  float* gWsp  = take((size_t)DD * DW);

  auto zero = [&](float* p, long n) {
    k_zero<<<dim3((unsigned)((n + 255) / 256)), dim3(256), 0, stream>>>(p, n);
  };
  auto padw = [&](const float* src, int rows, int cols, float* dst, int ldd) {
    long tot = (long)rows * ldd;
    k_padw<<<dim3((unsigned)((tot + 255) / 256)), dim3(256), 0, stream>>>(
        src, rows, cols, dst, ldd);
  };
  auto gemm = [&](const float* A, int lda, const float* W, int ldw,
                  const float* bias, float* O, int ldo, int M, int K, int n,
                  int flags, const int* gidx, const float* gsrc, int ldg) {
    dim3 g((unsigned)(ldw / 64), (unsigned)(M / 64));  // ldw == padded n
    k_gemm<<<g, dim3(128), 0, stream>>>(A, lda, W, ldw, bias, O, ldo, K, n,
                                        flags, gidx, gsrc, ldg);
  };

  const unsigned GND = (unsigned)(((long)NN * DD + 255) / 256);
  const unsigned GBD = (unsigned)(((long)BG * DD + 255) / 256);

  // ---- pad weights once ----
  padw(gine_W1, 5 * DD, D2, gW1p, D2P);
  padw(gine_W2, 5 * D2, DD, gW2p, DP);
  padw(gine_b1, 5, D2, gb1p, D2P);
  padw(gine_b2, 5, DD, gb2p, DP);
  padw(ce_W2, FDIM, DD, ceW2p, DW);
  padw(sage_Wl, DD, DD, sWlp, DW);
  padw(sage_Wr, DD, DD, sWrp, DW);
  padw(gat_Wsrc, DD, DD, gWsp, DW);

  // ---- embeddings ----
  k_embed<<<GND, 256, 0, stream>>>(xA, x_emb1, hA, NN, 2);
  k_embed<<<GND, 256, 0, stream>>>(xB, x_emb1, hB, NN, 2);

  // ---- context encoder: ctx = relu(context@ce_W1+b1)@ce_W2+b2 ----
  gemm(context, CDIM, ce_W1, FDIM, ce_b1, ceH, FDIM, BG, CDIM, FDIM,
       FL_RELU | FL_BIAS, nullptr, nullptr, 0);
  float* ctxCur = ctx0;
  float* ctxNxt = ctx1;
  gemm(ceH, FDIM, ceW2p, DW, ce_b2, ctxCur, DP, BG, FDIM, DD, FL_BIAS,
       nullptr, nullptr, 0);

  // constants
  k_wvec<<<dim3(5), dim3(64), 0, stream>>>(gat_Wdst, gat_adst, wvec);
  zero(cntA, BG); zero(cntB, BG);
  k_count<<<dim3((NN + 255) / 256), 256, 0, stream>>>(bA, cntA);
  k_count<<<dim3((NN + 255) / 256), 256, 0, stream>>>(bB, cntB);

  // ---- layers ----
  for (int l = 0; l < 5; ++l) {
    const float* W1 = gW1p + (size_t)l * DD * D2P;
    const float* b1 = gb1p + (size_t)l * D2P;
    const float* W2 = gW2p + (size_t)l * D2 * DP;
    const float* b2 = gb2p + (size_t)l * DP;
    const float* bg = bn_g + (size_t)l * DD;
    const float* bb = bn_b + (size_t)l * DD;
    const int relu = (l != 4);
    const int inject = (l >= 3);

    if (inject)
      gemm(ctxCur, DP, sWlp, DW, sage_bl, sageL, DP, BG, DD, DD, FL_BIAS,
           nullptr, nullptr, 0);

    for (int g = 0; g < 2; ++g) {
      float* h = (g == 0) ? hA : hB;
      const int* ei = (g == 0) ? eiA : eiB;
      const int* ea = (g == 0) ? eaA : eaB;
      const int* bt = (g == 0) ? bA : bB;

      float* hin = h;
      if (inject) {  // tmp1 = h @ Wr + sageL[batch]
        gemm(h, DP, sWrp, DW, nullptr, tmp1, DP, NN, DD, DD, FL_GATH, bt,
             sageL, DP);
        hin = tmp1;
      }
      zero(tmp2, (long)NP);  // agg
      k_scatter<<<dim3((EE * 75 + 255) / 256), 256, 0, stream>>>(
          ei, ea, hin, edge_emb1, tmp2);
      k_gine<<<dim3(NN / 16), dim3(256), 0, stream>>>(hin, tmp2, W1, b1, W2, b2,
                                                      hin);  // in-place
      zero(sumb, DP); zero(sumsq, DP);
      k_bnstats<<<dim3(256), dim3(256), 0, stream>>>(hin, sumb, sumsq, NN / 256);
      k_bnapply<<<GND, 256, 0, stream>>>(hin, sumb, sumsq, bg, bb, h, relu);
    }

    if (inject) {  // GAT readout on both graphs -> new ctx
      k_tctx<<<dim3(BG / 64), dim3(64), 0, stream>>>(ctxCur, wvec, tvec);
      zero(accA, (long)BP); zero(accB, (long)BP);
      for (int g = 0; g < 2; ++g) {
        const float* h = (g == 0) ? hA : hB;
        const int* bt = (g == 0) ? bA : bB;
        float* acc = (g == 0) ? accA : accB;
        gemm(h, DP, gWsp, DW, nullptr, tmp1, DP, NN, DD, DD, 0, nullptr,
             nullptr, 0);  // hs
        zero(den, BG);
        k_logits<<<dim3(NN / 8), dim3(256), 0, stream>>>(tmp1, gat_asrc, tvec,
                                                         bt, tmp2, den);
        k_alpha<<<dim3((NN + 255) / 256), 256, 0, stream>>>(tmp2, den, bt);
        k_gatacc<<<GND, 256, 0, stream>>>(tmp1, tmp2, bt, acc);
      }
      k_ctxcomb<<<GBD, 256, 0, stream>>>(accA, accB, gat_bias, ctxNxt);
      float* t = ctxCur; ctxCur = ctxNxt; ctxNxt = t;
    }
  }

  // ---- pooling + output layers ----
  for (int g = 0; g < 2; ++g) {
    const float* h = (g == 0) ? hA : hB;
    const int* bt = (g == 0) ? bA : bB;
    const float* cnt = (g == 0) ? cntA : cntB;
    float* pool = (g == 0) ? poolA : poolB;
    float* gout = (g == 0) ? gAo : gBo;
    zero(pool, (long)BP);
    k_poolacc<<<GND, 256, 0, stream>>>(h, bt, pool);
    k_pooldiv<<<GBD, 256, 0, stream>>>(pool, cnt);
    gemm(pool, DP, ol_W1, FDIM, ol_b1, ceH, FDIM, BG, DD, FDIM,
         FL_RELU | FL_BIAS, nullptr, nullptr, 0);
    gemm(ceH, FDIM, ol_W2, 256, ol_b2, gout, 256, BG, FDIM, 256, FL_BIAS,
         nullptr, nullptr, 0);
  }

  // ---- z = [gA, gB, ctx] ; head MLP ----
  k_copy2d<<<dim3((BG * 256 + 255) / 256), 256, 0, stream>>>(gAo, 256, zcat,
                                                             ZDIM, 0, 256);
  k_copy2d<<<dim3((BG * 256 + 255) / 256), 256, 0, stream>>>(gBo, 256, zcat,
                                                             ZDIM, 256, 256);
  k_copy2d<<<GBD, 256, 0, stream>>>(ctxCur, DP, zcat, ZDIM, 512, DD);

  gemm(zcat, ZDIM, mlp_W1, 256, mlp_b1, m1, 256, BG, ZDIM, 256,
       FL_RELU | FL_BIAS, nullptr, nullptr, 0);
  gemm(m1, 256, mlp_W2, 64, mlp_b2, m2, 64, BG, 256, 64, FL_RELU | FL_BIAS,
       nullptr, nullptr, 0);
  k_final<<<dim3(BG / 256), dim3(256), 0, stream>>>(m2, mlp_W3, mlp_b3,
                                                    (float*)d_out);
}